// SAGE_3298534884298
// MI455X (gfx1250) — compile-verified
//
#include <hip/hip_runtime.h>

#define N_NODES 50000
#define N_EDGES 1600000
#define HID 128
#define OUT_DIM 3
#define NEG_SLOPE 0.01f
#define KP 130  // padded LDS row stride in floats (K-major transposed weights)

typedef float v2f __attribute__((ext_vector_type(2)));
typedef float v4f __attribute__((ext_vector_type(4)));
typedef float v8f __attribute__((ext_vector_type(8)));

// ---------------------------------------------------------------- utilities
__global__ void fill_zero_kernel(float* __restrict__ p, int n) {
    int i = blockIdx.x * 256 + threadIdx.x;
    if (i < n) p[i] = 0.0f;
}

__global__ void count_edges_kernel(const int* __restrict__ dst,
                                   float* __restrict__ cnt) {
    int e = blockIdx.x * 256 + threadIdx.x;
    if (e < N_EDGES) atomicAdd(&cnt[dst[e]], 1.0f);
}

// One wave32 per edge: lane L moves floats [4L,4L+4) of the 128-float row.
__global__ void scatter_rows_kernel(const int* __restrict__ src,
                                    const int* __restrict__ dst,
                                    const float* __restrict__ X,
                                    float* __restrict__ S) {
    int e    = (blockIdx.x * 256 + threadIdx.x) >> 5;
    int lane = threadIdx.x & 31;
    if (e >= N_EDGES) return;
    int s = src[e];
    int d = dst[e];
    v4f v = *(const v4f*)(X + (size_t)s * HID + lane * 4);
    float* o = S + (size_t)d * HID + lane * 4;
    atomicAdd(o + 0, v.x);
    atomicAdd(o + 1, v.y);
    atomicAdd(o + 2, v.z);
    atomicAdd(o + 3, v.w);
}

// ---------------------------------------------------------- WMMA node GEMM
// out[M,128] = act( A1(*scale) @ W1  (+ A2 @ W2)  + bias )
//  - 256 threads = 8 wave32s; each wave computes a 16-row strip x 128 cols.
//  - fp32 WMMA 16x16x4; weights staged transposed (K-major) in LDS so the
//    B fragment is one aligned ds_load_b64; A fragment is one global b64.
//  - SCALE0: pass-0 A rows are scaled by 1/max(cnt,1) (SAGE mean-agg).
template <bool TWO_PASS, bool RELU, bool SCALE0>
__global__ __launch_bounds__(256) void gemm_node_kernel(
    const float* __restrict__ A1, const float* __restrict__ A2,
    const float* __restrict__ W1, const float* __restrict__ W2,
    const float* __restrict__ bias, const float* __restrict__ cnt,
    float* __restrict__ out) {
    __shared__ float lw[HID * KP];

    const int tid     = threadIdx.x;
    const int lane    = tid & 31;
    const int ln      = lane & 15;          // M (for A/C) or N (for B) sub-index
    const int koff    = (lane >> 4) * 2;    // lanes 16-31 carry K+2 (ISA layout)
    const int rowbase = blockIdx.x * 128 + (tid >> 5) * 16;

    const int arow   = rowbase + ln;
    const int arow_c = arow < N_NODES ? arow : (N_NODES - 1);  // keep EXEC full

    float inv = 1.0f;
    if (SCALE0) inv = 1.0f / fmaxf(cnt[arow_c], 1.0f);

    v8f acc[8];
#pragma unroll
    for (int nt = 0; nt < 8; ++nt) acc[nt] = (v8f){};

    const int npass = TWO_PASS ? 2 : 1;
    for (int pass = 0; pass < npass; ++pass) {
        // ---- stage W transposed into LDS: lw[n*KP + k] = W[k*HID + n]
        const float* W = (pass == 0) ? W1 : W2;
#pragma unroll
        for (int it = 0; it < 16; ++it) {
            int linear = it * 1024 + tid * 4;   // 256 thr * 4 floats
            int k  = linear >> 7;               // /128
            int n0 = linear & 127;
            v4f w = *(const v4f*)(W + linear);
            lw[(n0 + 0) * KP + k] = w.x;
            lw[(n0 + 1) * KP + k] = w.y;
            lw[(n0 + 2) * KP + k] = w.z;
            lw[(n0 + 3) * KP + k] = w.w;
        }
        __syncthreads();

        const float* A = (TWO_PASS && pass == 1) ? A2 : A1;
        const float* arowp = A + (size_t)arow_c * HID + koff;
        const float sc = (SCALE0 && pass == 0) ? inv : 1.0f;
        const float* lb = &lw[ln * KP + koff];

        if (TWO_PASS && pass == 0)
            __builtin_prefetch(A2 + (size_t)arow_c * HID + koff, 0, 1);

#pragma unroll 4
        for (int kk = 0; kk < 32; ++kk) {
            v2f a = *(const v2f*)(arowp + kk * 4);   // A frag: K=4kk+koff,+1
            a.x *= sc;
            a.y *= sc;
#pragma unroll
            for (int nt = 0; nt < 8; ++nt) {
                // B frag: W[4kk+koff .. +1][nt*16+ln]  (contiguous K-major)
                v2f b = *(const v2f*)(lb + nt * 16 * KP + kk * 4);
                acc[nt] = __builtin_amdgcn_wmma_f32_16x16x4_f32(
                    false, a, false, b, (short)0, acc[nt], false, false);
            }
        }
        __syncthreads();  // before next pass overwrites LDS
    }

    // ---- epilogue: bias (+ LeakyReLU), store per ISA C/D layout
    const int moff = (lane >> 4) * 8;  // lanes 16-31 hold rows M=8..15
#pragma unroll
    for (int nt = 0; nt < 8; ++nt) {
        float bv = bias[nt * 16 + ln];
#pragma unroll
        for (int r = 0; r < 8; ++r) {
            int row = rowbase + moff + r;
            if (row < N_NODES) {
                float v = acc[nt][r] + bv;
                if (RELU) v = v > 0.0f ? v : v * NEG_SLOPE;
                out[(size_t)row * HID + nt * 16 + ln] = v;
            }
        }
    }
}

// ------------------------------------------------------- final 128->3 proj
__global__ void out_proj_kernel(const float* __restrict__ H,
                                const float* __restrict__ Wo,
                                const float* __restrict__ bo,
                                float* __restrict__ out) {
    int t = blockIdx.x * 256 + threadIdx.x;
    if (t >= N_NODES * OUT_DIM) return;
    int n = t / OUT_DIM, j = t % OUT_DIM;
    const float* h = H + (size_t)n * HID;
    float s = bo[j];
#pragma unroll 8
    for (int k = 0; k < HID; ++k) s += h[k] * Wo[k * OUT_DIM + j];
    out[t] = s;
}

// ------------------------------------------------------------------ driver
extern "C" void kernel_launch(void* const* d_in, const int* in_sizes, int n_in,
                              void* d_out, int out_size, void* d_ws,
                              size_t ws_size, hipStream_t stream) {
    const float* feature = (const float*)d_in[0];
    const int*   eidx    = (const int*)d_in[1];
    const int*   src     = eidx;             // edge_index[0]
    const int*   dst     = eidx + N_EDGES;   // edge_index[1]
    // d_in[2] = edge_type (unused by reference math)
    const float* W_in  = (const float*)d_in[3];
    const float* b_in  = (const float*)d_in[4];
    const float* W1_l  = (const float*)d_in[5];
    const float* b1_l  = (const float*)d_in[6];
    const float* W1_r  = (const float*)d_in[7];
    const float* W2_l  = (const float*)d_in[8];
    const float* b2_l  = (const float*)d_in[9];
    const float* W2_r  = (const float*)d_in[10];
    const float* W_out = (const float*)d_in[11];
    const float* b_out = (const float*)d_in[12];

    const size_t NF = (size_t)N_NODES * HID;
    float* ws  = (float*)d_ws;
    float* X0  = ws;            // layer activations (reused for H2)
    float* S   = ws + NF;       // scatter-sum buffer
    float* H1  = ws + 2 * NF;   // conv1 output
    float* CNT = ws + 3 * NF;   // float in-degrees
    float* H2  = X0;            // conv2 output overwrites X0 (safe: X0 dead)

    const int nGemmWG = (N_NODES + 127) / 128;  // 391

    // degrees + zeroed scatter buffer
    fill_zero_kernel<<<(N_NODES + 255) / 256, 256, 0, stream>>>(CNT, N_NODES);
    fill_zero_kernel<<<(int)((NF + 255) / 256), 256, 0, stream>>>(S, (int)NF);
    count_edges_kernel<<<(N_EDGES + 255) / 256, 256, 0, stream>>>(dst, CNT);

    // X0 = leakyrelu(feature @ W_in + b_in)
    gemm_node_kernel<false, true, false><<<nGemmWG, 256, 0, stream>>>(
        feature, nullptr, W_in, nullptr, b_in, nullptr, X0);

    // conv1: S = segment_sum(X0[src] -> dst); H1 = (S/cnt)@W1_l + X0@W1_r + b1_l
    scatter_rows_kernel<<<N_EDGES / 8, 256, 0, stream>>>(src, dst, X0, S);
    gemm_node_kernel<true, false, true><<<nGemmWG, 256, 0, stream>>>(
        S, X0, W1_l, W1_r, b1_l, CNT, H1);

    // conv2: re-zero S, scatter H1, fused GEMM -> H2
    fill_zero_kernel<<<(int)((NF + 255) / 256), 256, 0, stream>>>(S, (int)NF);
    scatter_rows_kernel<<<N_EDGES / 8, 256, 0, stream>>>(src, dst, H1, S);
    gemm_node_kernel<true, false, true><<<nGemmWG, 256, 0, stream>>>(
        S, H1, W2_l, W2_r, b2_l, CNT, H2);

    // final projection
    out_proj_kernel<<<(N_NODES * OUT_DIM + 255) / 256, 256, 0, stream>>>(
        H2, W_out, b_out, (float*)d_out);
}